// MultiCamVoxelSampler_90074054132502
// MI455X (gfx1250) — compile-verified
//
#include <hip/hip_runtime.h>
#include <stdint.h>

// MultiCamVoxelSampler for MI455X (gfx1250).
// Gather/store-bound: 51.2MB output (~2.2us floor @23.3TB/s), 10.8MB features
// are L2-resident. CDNA5 path: double-buffered TDM tensor_load_to_lds pipeline
// (depth-1, s_wait_tensorcnt 0x1) overlaps each camera tile's DMA with the
// previous camera's LDS bilinear gathers.

#define NCAM  6
#define CTOT  80
#define FH    32
#define FW    88
#define HWSZ  (FH * FW)          // 2816 floats per channel plane
#define CH    2                  // channels per TDM tile (2*2816*4B = 22528B)
#define NVOX  80000              // 8 * 100 * 100
#define NYG   100
#define NXG   100
#define TILE_ELEMS (CH * HWSZ)   // 5632 4-byte elements per TDM tile

typedef uint32_t u32x4 __attribute__((ext_vector_type(4)));
typedef uint32_t u32x8 __attribute__((ext_vector_type(8)));

__device__ __forceinline__ void tdm_load_tile(const float* gbase, float* lbase)
{
    const uint64_t ga = (uint64_t)(uintptr_t)gbase;
    const uint32_t la = (uint32_t)(uintptr_t)lbase;
    // D# group0: count=1 | lds_addr | global_addr[56:0] | type=2
    u32x4 g0;
    g0.x = 1u;
    g0.y = la;
    g0.z = (uint32_t)ga;
    g0.w = ((uint32_t)(ga >> 32) & 0x01FFFFFFu) | (2u << 30);
    // D# group1: data_size=2(4B); tensor_dim0=tile_dim0=5632; tensor_dim1=1;
    // tile_dim1/2=0 (1-D tile); stride0=5632; no pad/iterate/multicast.
    u32x8 g1;
    g1.s0 = 0x00020000u;                              // data_size=2 @[17:16]
    g1.s1 = (uint32_t)(TILE_ELEMS & 0xFFFF) << 16;    // tensor_dim0 lo16 @[63:48]
    g1.s2 = 0x00010000u;                              // tensor_dim1=1 @[111:80]
    g1.s3 = (uint32_t)(TILE_ELEMS & 0xFFFF) << 16;    // tile_dim0 @[127:112]
    g1.s4 = 0u;                                       // tile_dim1=0, tile_dim2=0
    g1.s5 = (uint32_t)TILE_ELEMS;                     // tensor_dim0_stride lo32
    g1.s6 = 0u;
    g1.s7 = 0u;
    u32x4 gz = {0u, 0u, 0u, 0u};                      // groups 2/3: dims unused
    asm volatile("tensor_load_to_lds %0, %1, %2, %3"
                 :: "s"(g0), "s"(g1), "s"(gz), "s"(gz) : "memory");
}

__global__ __launch_bounds__(256)
void voxsampler_kernel(const float* __restrict__ feats,
                       const float* __restrict__ Ks,
                       const float* __restrict__ s2e,
                       float* __restrict__ out)
{
    __shared__ float smem[2 * TILE_ELEMS];            // 45056B double buffer

    const int tid = threadIdx.x;
    const int b   = blockIdx.z;
    const int c0  = blockIdx.y * CH;
    const int p   = blockIdx.x * 256 + tid;
    const bool pvalid = (p < NVOX);

    // Voxel center: p = (z*100 + x)*100 + y ; centers at step-1 grid midpoints.
    const int   iy  = p % NYG;
    const int   tq  = p / NYG;
    const int   ixg = tq % NXG;
    const int   izg = tq / NXG;
    const float px  = (float)ixg - 49.5f;
    const float py  = (float)iy  - 49.5f;
    const float pz  = (float)izg - 4.5f;

    const float* fbase = feats + (size_t)((b * NCAM) * CTOT + c0) * HWSZ;
    const size_t camstride = (size_t)CTOT * HWSZ;     // floats between cameras

    float acc0 = 0.f, acc1 = 0.f;
    float cnt  = 0.f;

    // Prologue: wave 0 kicks off the DMA for camera 0 into buffer 0.
    if (tid < 32) {
        tdm_load_tile(fbase, &smem[0]);
    }

    for (int n = 0; n < NCAM; ++n) {
        // Wave 0: issue next camera's DMA into the other buffer, then wait for
        // the current camera's DMA (in-order completion => tensorcnt<=1).
        if (tid < 32) {
            if (n < NCAM - 1) {
                tdm_load_tile(fbase + (size_t)(n + 1) * camstride,
                              &smem[((n + 1) & 1) * TILE_ELEMS]);
                __builtin_amdgcn_s_wait_tensorcnt(1);
            } else {
                __builtin_amdgcn_s_wait_tensorcnt(0);
            }
        }
        __syncthreads();                              // tile n visible to all waves

        // ---- Geometry (K, R, t wave-uniform per (b,n) -> scalar loads).
        const float* K = Ks  + (size_t)((b * NCAM) + n) * 9;
        const float* T = s2e + (size_t)((b * NCAM) + n) * 16;
        const float fx = K[0], fy = K[4], cx = K[2], cy = K[5];
        const float dx = px - T[3], dy = py - T[7], dz = pz - T[11];
        const float Xc = T[0] * dx + T[1] * dy + T[2]  * dz;
        const float Yc = T[4] * dx + T[5] * dy + T[6]  * dz;
        const float Zc = T[8] * dx + T[9] * dy + T[10] * dz;
        const float Zs = fmaxf(Zc, 1e-4f);
        const float u  = Xc * fx / Zs + cx;
        const float v  = Yc * fy / Zs + cy;
        const float un = u / (float)(FW - 1) * 2.0f - 1.0f;
        const float vn = v / (float)(FH - 1) * 2.0f - 1.0f;
        const bool valid = pvalid && (Zc > 1e-4f) &&
                           (un >= -1.0f) && (un <= 1.0f) &&
                           (vn >= -1.0f) && (vn <= 1.0f);
        if (valid) {
            const float xg  = (un + 1.0f) * 0.5f * (float)(FW - 1);
            const float yg  = (vn + 1.0f) * 0.5f * (float)(FH - 1);
            const float x0f = floorf(xg), y0f = floorf(yg);
            const float wx1 = xg - x0f,   wy1 = yg - y0f;
            const float wx0 = 1.0f - wx1, wy0 = 1.0f - wy1;
            // valid => xg in [0,87], yg in [0,31]; OOB corners carry weight 0,
            // so clamping reproduces the reference's per-corner bounds mask.
            const int x0 = min(max((int)x0f, 0), FW - 1);
            const int y0 = min(max((int)y0f, 0), FH - 1);
            const int x1 = min(x0 + 1, FW - 1);
            const int y1 = min(y0 + 1, FH - 1);
            const int i00 = y0 * FW + x0, i01 = y0 * FW + x1;
            const int i10 = y1 * FW + x0, i11 = y1 * FW + x1;
            const float w00 = wx0 * wy0, w01 = wx1 * wy0;
            const float w10 = wx0 * wy1, w11 = wx1 * wy1;

            const float* s = &smem[(n & 1) * TILE_ELEMS];
            acc0 += w00 * s[i00] + w01 * s[i01] + w10 * s[i10] + w11 * s[i11]; s += HWSZ;
            acc1 += w00 * s[i00] + w01 * s[i01] + w10 * s[i10] + w11 * s[i11];
            cnt  += 1.0f;
        }
        __syncthreads();   // buffer (n&1) free for DMA of camera n+2
    }

    if (pvalid) {
        const float inv = 1.0f / fmaxf(cnt, 1.0f);
        size_t ob = ((size_t)b * CTOT + c0) * (size_t)NVOX + (size_t)p;
        out[ob]               = acc0 * inv;
        out[ob + (size_t)NVOX] = acc1 * inv;
    }
}

extern "C" void kernel_launch(void* const* d_in, const int* in_sizes, int n_in,
                              void* d_out, int out_size, void* d_ws, size_t ws_size,
                              hipStream_t stream) {
    (void)in_sizes; (void)n_in; (void)d_ws; (void)ws_size; (void)out_size;
    const float* feats = (const float*)d_in[0];   // (B,N,C,H,W) f32
    const float* Ks    = (const float*)d_in[1];   // (B,N,3,3)  f32
    const float* s2e   = (const float*)d_in[2];   // (B,N,4,4)  f32
    float* out = (float*)d_out;                   // (B,C,8,100,100) f32

    dim3 grid((NVOX + 255) / 256, CTOT / CH, 2);
    voxsampler_kernel<<<grid, dim3(256, 1, 1), 0, stream>>>(feats, Ks, s2e, out);
}